// LagrKANNautoinner_11055245820076
// MI455X (gfx1250) — compile-verified
//
#include <hip/hip_runtime.h>

// ---------------------------------------------------------------------------
// LagrKAN forward for MI455X (gfx1250, wave32).
// Outputs (flat, concatenated): t(4096x16), dt, ddt, phi(4096x16x193x2),
// dphi, ddphi, delta_x (scalar).
// Memory-bound: ~304 MB of output -> nontemporal B128 streaming stores.
// Einsums done as dense-K f32 GEMM tiles via V_WMMA_F32_16X16X4_F32.
// ---------------------------------------------------------------------------

typedef __attribute__((ext_vector_type(2))) float v2f;
typedef __attribute__((ext_vector_type(4))) float f4;
typedef __attribute__((ext_vector_type(8))) float v8f;

#define NSAMP   4096
#define NWIDTH  16
#define NNODES  193
#define COLS    386            // NNODES * 2
#define CHUNK   6176           // NWIDTH * COLS floats per sample per big array
#define T_OFF   0
#define DT_OFF  65536
#define DDT_OFF 131072
#define PHI_OFF 196608
#define BIG     25296896       // 4096*16*193*2
#define DX_OFF  (PHI_OFF + 3 * BIG)   // 76087296

// Lagrange basis (P=4, nodes linspace(-1,1,4)): value / d/dx / d2/dx2.
// Division-by-constant folded to multiply-by-reciprocal at compile time.
__device__ __forceinline__ void lagrange4(float xr, float* phi, float* dphi,
                                          float* ddphi) {
  constexpr float nd[4] = {-1.0f, -1.0f / 3.0f, 1.0f / 3.0f, 1.0f};
#pragma unroll
  for (int j = 0; j < 4; ++j) {
    float pv = 1.0f;
#pragma unroll
    for (int m = 0; m < 4; ++m)
      if (m != j) pv *= (xr - nd[m]) * (1.0f / (nd[j] - nd[m]));
    phi[j] = pv;

    float y1 = 0.0f;
#pragma unroll
    for (int i = 0; i < 4; ++i) {
      if (i == j) continue;
      float k = 1.0f / (nd[j] - nd[i]);
#pragma unroll
      for (int m = 0; m < 4; ++m)
        if (m != i && m != j) k *= (xr - nd[m]) * (1.0f / (nd[j] - nd[m]));
      y1 += k;
    }
    dphi[j] = y1;

    float y2 = 0.0f;
#pragma unroll
    for (int i = 0; i < 4; ++i) {
      if (i == j) continue;
      float ks = 0.0f;
#pragma unroll
      for (int m = 0; m < 4; ++m) {
        if (m == i || m == j) continue;
        float kp = 1.0f / (nd[j] - nd[m]);
#pragma unroll
        for (int n = 0; n < 4; ++n)
          if (n != i && n != j && n != m)
            kp *= (xr - nd[n]) * (1.0f / (nd[j] - nd[n]));
        ks += kp;
      }
      y2 += ks * (1.0f / (nd[j] - nd[i]));
    }
    ddphi[j] = y2;
  }
}

// Full pointwise pipeline for one (sample, dim): element lookup + scaled basis.
// 1/delta_x = 128 and 1/delta_x^2 = 16384 are exact powers of two.
__device__ __forceinline__ void basis_eval(float xv, int& nl, float* ph,
                                           float* dp, float* dd) {
  float x_shift = 192.0f * xv;                 // (N_NODES-1) * (x - 0) / 1
  float fel = floorf(x_shift / 3.0f);          // match reference's fp divide
  fel = fminf(fmaxf(fel, 0.0f), 63.0f);
  nl = (int)fel * 3;
  float xr = (x_shift - (float)nl) * (2.0f / 3.0f) - 1.0f;
  lagrange4(xr, ph, dp, dd);
#pragma unroll
  for (int p = 0; p < 4; ++p) {
    dp[p] *= 128.0f;
    dd[p] *= 16384.0f;
  }
}

// ---------------------------------------------------------------------------
// Kernel 1: t/dt/ddt via f32 WMMA. One wave per 16-sample tile; N = all 16
// widths. Dense-K loop over the 386 (node,dim) columns in steps of 4.
// ---------------------------------------------------------------------------
__global__ void __launch_bounds__(32)
lagkan_einsum_wmma(const float* __restrict__ x, const float* __restrict__ w,
                   float* __restrict__ out) {
  __shared__ __align__(16) float s_w[CHUNK];   // weight as (k, c=n*2+j), 24.7 KB
  __shared__ int s_nl[16][2];                  // left node index per (m, j)
  __shared__ float s_tab[3][16][4][2];         // [phi|dphi|ddphi][m][p][j]

  const int lane = threadIdx.x;
  const int i0 = blockIdx.x * 16;

  // Stage weights into LDS with 16B loads.
  for (int idx = lane; idx < CHUNK / 4; idx += 32)
    ((f4*)s_w)[idx] = ((const f4*)w)[idx];

  // 32 lanes = 16 samples x 2 dims: evaluate basis once per (i, j).
  {
    const int il = lane >> 1, j = lane & 1;
    int nl;
    float ph[4], dp[4], dd[4];
    basis_eval(x[(i0 + il) * 2 + j], nl, ph, dp, dd);
    s_nl[il][j] = nl;
#pragma unroll
    for (int p = 0; p < 4; ++p) {
      s_tab[0][il][p][j] = ph[p];
      s_tab[1][il][p][j] = dp[p];
      s_tab[2][il][p][j] = dd[p];
    }
  }
  __syncthreads();

  // A (16x4 f32): lanes 0-15 hold row M=lane, K slots {0,1}; lanes 16-31 hold
  // row M=lane-16, K slots {2,3}. B (4x16): lane's column N=lane&15, same K
  // slot split. Branchless gathers keep EXEC all-ones around WMMA.
  const int MN = lane & 15;
  const int koff = (lane >> 4) << 1;

  auto acol = [&](int tsel, int c) -> float {
    int n = c >> 1, j = c & 1;
    int p = n - s_nl[MN][j];
    float v = s_tab[tsel][MN][p & 3][j];       // clamped (in-range) LDS read
    return ((unsigned)p < 4u) ? v : 0.0f;      // v_cndmask, no branch
  };
  auto bcol = [&](int c) -> float {
    int cc = (c < COLS) ? c : 0;
    float v = s_w[MN * COLS + cc];
    return (c < COLS) ? v : 0.0f;
  };

  v8f acc_t = {};
  v8f acc_d = {};
  v8f acc_dd = {};
  for (int Kb = 0; Kb < 388; Kb += 4) {        // 97 iterations, cols 0..387
    const int c0 = Kb + koff, c1 = c0 + 1;
    v2f ap, ad, a2, b;
    ap.x = acol(0, c0);  ap.y = acol(0, c1);
    ad.x = acol(1, c0);  ad.y = acol(1, c1);
    a2.x = acol(2, c0);  a2.y = acol(2, c1);
    b.x  = bcol(c0);     b.y  = bcol(c1);
    acc_t  = __builtin_amdgcn_wmma_f32_16x16x4_f32(false, ap, false, b,
                                                   (short)0, acc_t, false, false);
    acc_d  = __builtin_amdgcn_wmma_f32_16x16x4_f32(false, ad, false, b,
                                                   (short)0, acc_d, false, false);
    acc_dd = __builtin_amdgcn_wmma_f32_16x16x4_f32(false, a2, false, b,
                                                   (short)0, acc_dd, false, false);
  }

  // C/D 16x16 f32 layout: VGPR r -> M = r (lanes 0-15) or 8+r (lanes 16-31),
  // N = lane & 15.
#pragma unroll
  for (int r = 0; r < 8; ++r) {
    const int row = i0 + r + ((lane >> 4) << 3);
    const int o = row * NWIDTH + MN;
    out[T_OFF + o]   = acc_t[r];
    out[DT_OFF + o]  = acc_d[r];
    out[DDT_OFF + o] = acc_dd[r];
  }

  if (blockIdx.x == 0 && lane == 0) out[DX_OFF] = 0.0078125f;  // delta_x
}

// ---------------------------------------------------------------------------
// Kernel 2: stream the three big (4096,16,193,2) arrays. The 6176-float
// per-sample chunk is 16 repeats of one 386-float row (independent of k),
// so build the row once in LDS and replay it with NT float4 stores.
// ---------------------------------------------------------------------------
__global__ void __launch_bounds__(256)
lagkan_fill(const float* __restrict__ x, float* __restrict__ out) {
  __shared__ int s_nl[2];
  __shared__ float s_tab[3][4][2];
  __shared__ float s_row[3][388];

  const int tid = threadIdx.x;
  const int i = blockIdx.x;

  if (tid < 2) {
    const int j = tid;
    int nl;
    float ph[4], dp[4], dd[4];
    basis_eval(x[i * 2 + j], nl, ph, dp, dd);
    s_nl[j] = nl;
#pragma unroll
    for (int p = 0; p < 4; ++p) {
      s_tab[0][p][j] = ph[p];
      s_tab[1][p][j] = dp[p];
      s_tab[2][p][j] = dd[p];
    }
  }
  __syncthreads();

  for (int idx = tid; idx < 3 * COLS; idx += 256) {
    const int a = idx / COLS, c = idx - a * COLS;
    const int n = c >> 1, j = c & 1;
    const int p = n - s_nl[j];
    s_row[a][c] = ((unsigned)p < 4u) ? s_tab[a][p & 3][j] : 0.0f;
  }
  __syncthreads();

  float* const base0 = out + (size_t)PHI_OFF + (size_t)i * CHUNK;
#pragma unroll
  for (int a = 0; a < 3; ++a) {
    float* const bp = base0 + (size_t)a * BIG;
    const float* const row = s_row[a];
    for (int idx = tid; idx < CHUNK / 4; idx += 256) {
      const int e0 = idx << 2;
      int c = e0 % COLS;
      f4 v;
      v.x = row[c]; c = (c + 1 == COLS) ? 0 : c + 1;
      v.y = row[c]; c = (c + 1 == COLS) ? 0 : c + 1;
      v.z = row[c]; c = (c + 1 == COLS) ? 0 : c + 1;
      v.w = row[c];
      __builtin_nontemporal_store(v, (f4*)(bp + e0));  // ~304 MB total, NT
    }
  }
}

extern "C" void kernel_launch(void* const* d_in, const int* in_sizes, int n_in,
                              void* d_out, int out_size, void* d_ws,
                              size_t ws_size, hipStream_t stream) {
  (void)in_sizes; (void)n_in; (void)out_size; (void)d_ws; (void)ws_size;
  const float* x = (const float*)d_in[0];     // (4096, 2)
  const float* w = (const float*)d_in[1];     // (16, 193, 2)
  float* out = (float*)d_out;

  lagkan_einsum_wmma<<<NSAMP / 16, 32, 0, stream>>>(x, w, out);
  lagkan_fill<<<NSAMP, 256, 0, stream>>>(x, out);
}